// QuadTree_62732292325557
// MI455X (gfx1250) — compile-verified
//
#include <hip/hip_runtime.h>

#define NMAT 2048
#define SZ   128
#define EMB  64

typedef float v2f __attribute__((ext_vector_type(2)));
typedef float v8f __attribute__((ext_vector_type(8)));

__device__ __forceinline__ unsigned part1by1(unsigned x) {
    x &= 0xFFu;
    x = (x | (x << 4)) & 0x0F0Fu;
    x = (x | (x << 2)) & 0x3333u;
    x = (x | (x << 1)) & 0x5555u;
    return x;
}

// LDS byte offsets per quadtree level (level 7 padded to 4B alignment for packed stores)
#define LOFF(l) ((l) == 7 ? 5464 : (((1 << (2 * (l))) - 1) / 3))
// Node-id offsets in E (true quadtree offsets)
#define QOFF(l) (((1 << (2 * (l))) - 1) / 3)

// Kernel 1: per-matrix quadtree reduction -> selected weights -> A row (L1-normalized)
__global__ __launch_bounds__(256) void qt_weights_A(const int* __restrict__ mats,
                                                    const float* __restrict__ E,
                                                    float* __restrict__ A) {
    __shared__ unsigned char lvl[5464 + SZ * SZ];   // 21848 bytes of quadtree flags
    __shared__ float part[4][EMB];

    const int b   = blockIdx.x;
    const int tid = threadIdx.x;

    // --- leaf level (l=7): mask bytes, packed 4 per 32-bit LDS store ---
    const int4* m4 = (const int4*)(mats + (size_t)b * (SZ * SZ));
    unsigned*   L7 = (unsigned*)(lvl + LOFF(7));
    for (int i = tid; i < (SZ * SZ) / 4; i += 256) {
        int4 v = m4[i];
        L7[i] = (v.x != 0 ? 0x00000001u : 0u) |
                (v.y != 0 ? 0x00000100u : 0u) |
                (v.z != 0 ? 0x00010000u : 0u) |
                (v.w != 0 ? 0x01000000u : 0u);
    }
    __syncthreads();

    // --- AND-reduce levels 6..0 ---
#pragma unroll
    for (int l = 6; l >= 0; --l) {
        const int n  = 1 << l;
        const int n2 = n << 1;
        unsigned char*       dst = lvl + LOFF(l);
        const unsigned char* src = lvl + LOFF(l + 1);
        for (int idx = tid; idx < n * n; idx += 256) {
            const int gy = idx >> l;
            const int gx = idx & (n - 1);
            const int base = (gy << 1) * n2 + (gx << 1);
            dst[idx] = (unsigned char)(src[base] & src[base + 1] &
                                       src[base + n2] & src[base + n2 + 1]);
        }
        __syncthreads();
    }

    // --- accumulate A[b] = sum over selected nodes of coef * E[node] ---
    const int e = tid & (EMB - 1);   // embedding column
    const int g = tid >> 6;          // group 0..3 (64 threads each; coherent per wave32)
    float sum = 0.0f;
#pragma unroll
    for (int l = 0; l < 8; ++l) {
        const int n  = 1 << l;
        const int nn = n * n;
        const unsigned char* cur = lvl + LOFF(l);
        const unsigned char* par = (l == 0) ? cur : lvl + LOFF(l - 1);
        const float coef = 1.0f / (float)(1 << (2 * l));
        const int   noff = QOFF(l);
        for (int idx = g; idx < nn; idx += 4) {
            const int gy = idx >> l;
            const int gx = idx & (n - 1);
            bool sel;
            if (l == 0) {
                sel = cur[0] != 0;
            } else {
                const int pidx = (gy >> 1) * (n >> 1) + (gx >> 1);
                sel = (cur[idx] != 0) && (par[pidx] == 0);
            }
            if (sel) {
                const unsigned loc = part1by1((unsigned)gx) | (part1by1((unsigned)gy) << 1);
                sum += coef * E[(size_t)(noff + (int)loc) * EMB + e];
            }
        }
    }
    part[g][e] = sum;
    __syncthreads();

    // combine 4 groups + row L1 norm
    if (tid < EMB) {
        const float v = part[0][tid] + part[1][tid] + part[2][tid] + part[3][tid];
        part[0][tid] = v;
        part[1][tid] = fabsf(v);
    }
    __syncthreads();
#pragma unroll
    for (int s = 32; s >= 1; s >>= 1) {
        if (tid < s) part[1][tid] += part[1][tid + s];
        __syncthreads();
    }
    if (tid < EMB) {
        A[(size_t)b * EMB + tid] = part[0][tid] / part[1][0];
    }
}

// Kernel 2: G = A * A^T via V_WMMA_F32_16X16X4_F32 (full f32), plus |G| column partials.
// 8 waves per block, one 16x16 tile per wave.
__global__ __launch_bounds__(256) void qt_gram_wmma(const float* __restrict__ A,
                                                    float* __restrict__ G,
                                                    float* __restrict__ colsum) {
    const int tile = blockIdx.x * 8 + (threadIdx.x >> 5);
    const int I    = tile >> 7;          // row tile 0..127
    const int J    = tile & 127;         // col tile 0..127
    const int lane = threadIdx.x & 31;
    const int lo   = lane & 15;
    const int koff = (lane >> 4) << 1;   // 0 for lanes 0-15, 2 for lanes 16-31

    const float* arow = A + (size_t)(I * 16 + lo) * EMB + koff;  // A-tile row, this lane
    const float* brow = A + (size_t)(J * 16 + lo) * EMB + koff;  // B = A_J^T, same pattern

    v8f c = {};
#if __has_builtin(__builtin_amdgcn_wmma_f32_16x16x4_f32)
#pragma unroll
    for (int kk = 0; kk < EMB / 4; ++kk) {
        const v2f a = *(const v2f*)(arow + kk * 4);   // K = 4kk+koff, 4kk+koff+1
        const v2f bm = *(const v2f*)(brow + kk * 4);
        c = __builtin_amdgcn_wmma_f32_16x16x4_f32(false, a, false, bm,
                                                  (short)0, c, false, false);
    }
#else
    // scalar fallback producing the same C/D lane layout
    {
        const int mrow = I * 16 + ((lane >> 4) << 3);
#pragma unroll
        for (int r = 0; r < 8; ++r) {
            float acc = 0.0f;
            for (int k = 0; k < EMB; ++k)
                acc += A[(size_t)(mrow + r) * EMB + k] * A[(size_t)(J * 16 + lo) * EMB + k];
            c[r] = acc;
        }
    }
#endif

    // C/D layout: VGPR r -> (M = r + 8*(lane>=16), N = lane&15)
    const int nCol  = J * 16 + lo;
    const int rbase = I * 16 + ((lane >> 4) << 3);
    float s = 0.0f;
#pragma unroll
    for (int r = 0; r < 8; ++r) {
        const float v = c[r];
        G[(size_t)(rbase + r) * 2048 + nCol] = v;
        s += fabsf(v);
    }
    atomicAdd(&colsum[nCol], s);   // two partials per column per tile (lane halves)
}

// Kernel 3: in-place column normalization of G
__global__ __launch_bounds__(256) void qt_colnorm(float* __restrict__ G,
                                                  const float* __restrict__ colsum) {
    const int idx = blockIdx.x * 256 + threadIdx.x;   // 4M elements
    const int j   = idx & 2047;
    G[idx] = G[idx] / colsum[j];
}

extern "C" void kernel_launch(void* const* d_in, const int* in_sizes, int n_in,
                              void* d_out, int out_size, void* d_ws, size_t ws_size,
                              hipStream_t stream) {
    (void)in_sizes; (void)n_in; (void)out_size; (void)ws_size;
    const int*   mats = (const int*)d_in[0];
    const float* E    = (const float*)d_in[1];
    float*       G    = (float*)d_out;

    float* A      = (float*)d_ws;                                   // 2048*64 f32 = 512 KB
    float* colsum = (float*)((char*)d_ws + (size_t)NMAT * EMB * 4); // 2048 f32

    hipMemsetAsync(colsum, 0, NMAT * sizeof(float), stream);
    qt_weights_A<<<dim3(NMAT), dim3(256), 0, stream>>>(mats, E, A);
    qt_gram_wmma<<<dim3((128 * 128) / 8), dim3(256), 0, stream>>>(A, G, colsum);
    qt_colnorm<<<dim3((NMAT * NMAT) / 256), dim3(256), 0, stream>>>(G, colsum);
}